// Wav2Vec2Mask_74878459838966
// MI455X (gfx1250) — compile-verified
//
#include <hip/hip_runtime.h>

// ---- problem constants (match reference) ----
#define N_B      16
#define C_DIM    512
#define W_DIM    4096
#define LOG2W    12
#define F_DIM    512
#define K_DIM    512
#define T_MASKED 1024                    // NUM_MASKS(256) * MASK_WIDTH(4)
#define T_NEG    10240                   // 10 * 256 * 4
#define M_ROWS   (T_MASKED + T_NEG)      // 11264
#define OUT0_ELEMS (N_B * C_DIM * W_DIM) // 33554432 floats
#define LDS_STRIDE 516                   // 512 + 4 pad: kill LDS bank conflicts

typedef __attribute__((ext_vector_type(2))) float v2f;
typedef __attribute__((ext_vector_type(8))) float v8f;

// ---------------------------------------------------------------------------
// Kernel 1: out[b,c,0,t] = (t < seq_len[b]) ? inputs[b,c,0,t] : 0
// One block per (b,c) row; coalesced float4 stream along W.
// ---------------------------------------------------------------------------
__global__ void __launch_bounds__(256)
k_copy_pad(const float* __restrict__ in, const int* __restrict__ seq_len,
           float* __restrict__ out) {
    const int row = blockIdx.x;              // 0 .. N_B*C_DIM-1
    const int b   = row >> 9;                // / C_DIM
    const int len = seq_len[b];
    const float4* src = reinterpret_cast<const float4*>(in  + (size_t)row * W_DIM);
    float4*       dst = reinterpret_cast<float4*>      (out + (size_t)row * W_DIM);
    const int tid = threadIdx.x;
#pragma unroll
    for (int j = 0; j < 4; ++j) {
        const int t4 = tid + j * 256;        // float4 index, 1024 per row
        float4 v = src[t4];
        const int t = t4 << 2;
        v.x = (t + 0 < len) ? v.x : 0.0f;
        v.y = (t + 1 < len) ? v.y : 0.0f;
        v.z = (t + 2 < len) ? v.z : 0.0f;
        v.w = (t + 3 < len) ? v.w : 0.0f;
        dst[t4] = v;
    }
}

// ---------------------------------------------------------------------------
// Kernel 2: overwrite the 1024 masked (b,t) positions with tiled mask_emb.
// out[b, c, 0, t] = mask_emb[k % 4, c]   where flat = pack_idx[masked_idx[k]]
// One block per masked token; stride-W scatter along c (small traffic).
// ---------------------------------------------------------------------------
__global__ void __launch_bounds__(256)
k_scatter_mask(const int* __restrict__ pack_idx, const int* __restrict__ masked_idx,
               const float* __restrict__ mask_emb, float* __restrict__ out) {
    const int k    = blockIdx.x;                     // 0..1023
    const int flat = pack_idx[masked_idx[k]];        // b*W + t
    const int b    = flat >> LOG2W;
    const int t    = flat & (W_DIM - 1);
    const float* emb = mask_emb + (size_t)(k & 3) * C_DIM;
    float* dst = out + (size_t)b * (C_DIM * (size_t)W_DIM) + t;
#pragma unroll
    for (int c = threadIdx.x; c < C_DIM; c += 256)
        dst[(size_t)c * W_DIM] = emb[c];
}

// ---------------------------------------------------------------------------
// Kernel 3: fused gather-GEMM, fp32 via V_WMMA_F32_16X16X4_F32.
//   rows r in [0,11264): src token = pack_idx[ r<1024 ? masked_idx[r]
//                                                    : neg_idx[r-1024] ]
//   Out[r, f] = sum_c inputs[b, c, 0, t] * Wq[f, c] + bq[f]
// Block: 256 threads = 8 waves. Each wave -> one 16x16 output tile.
// Grid: (M_ROWS/16, F_DIM/128). A tile (16 x 512) staged in LDS per block.
// ---------------------------------------------------------------------------
__global__ void __launch_bounds__(256)
k_proj_gemm(const float* __restrict__ in, const int* __restrict__ pack_idx,
            const int* __restrict__ masked_idx, const int* __restrict__ neg_idx,
            const float* __restrict__ Wq, const float* __restrict__ bq,
            float* __restrict__ out) {
    __shared__ float As[16 * LDS_STRIDE];   // 33 KB
    __shared__ int   rowBase[16];

    const int tid  = threadIdx.x;
    const int m0   = blockIdx.x << 4;                 // first of 16 rows
    const int f0   = (blockIdx.y << 7) + ((tid >> 5) << 4); // wave's 16 F-cols

    // --- resolve the 16 gathered row base addresses ---
    if (tid < 16) {
        const int r    = m0 + tid;
        const int sel  = (r < T_MASKED) ? masked_idx[r] : neg_idx[r - T_MASKED];
        const int flat = pack_idx[sel];               // b*W + t
        const int b    = flat >> LOG2W;
        const int t    = flat & (W_DIM - 1);
        rowBase[tid]   = b * (C_DIM * W_DIM) + t;     // channel c adds c*W
    }
    __syncthreads();

    // --- gather A tile: As[i][c] = inputs[base_i + c*W] ---
#pragma unroll
    for (int e = tid; e < 16 * K_DIM; e += 256) {
        const int i = e >> 9;          // row 0..15
        const int c = e & (K_DIM - 1); // channel
        As[i * LDS_STRIDE + c] = in[(size_t)rowBase[i] + (size_t)c * W_DIM];
    }
    __syncthreads();

    // --- WMMA main loop: D(16x16) += A(16x4) x B(4x16), 128 k-steps ---
    const int lane = tid & 31;
    const int lo16 = lane & 15;
    const int hi   = lane >> 4;        // 0: K pair {0,1}, 1: K pair {2,3}

    v8f acc = {0.f, 0.f, 0.f, 0.f, 0.f, 0.f, 0.f, 0.f};
    const float* wrow = Wq + (size_t)(f0 + lo16) * K_DIM;  // B[n] row, contiguous k

#pragma unroll 8
    for (int k0 = 0; k0 < K_DIM; k0 += 4) {
        // A frag: lane<16 -> A[M=lane, k0..k0+1]; lane>=16 -> A[M, k0+2..k0+3]
        v2f a = *reinterpret_cast<const v2f*>(&As[lo16 * LDS_STRIDE + k0 + 2 * hi]);
        // B frag: B[k, n] = Wq[f0+n, k]; same K striping as A
        v2f b = *reinterpret_cast<const v2f*>(&wrow[k0 + 2 * hi]);
        acc = __builtin_amdgcn_wmma_f32_16x16x4_f32(
            /*neg_a=*/false, a, /*neg_b=*/false, b,
            /*c_mod=*/(short)0, acc, /*reuse_a=*/false, /*reuse_b=*/false);
    }

    // --- epilogue: bias + store. VGPR v: M=v (lanes 0-15), M=8+v (lanes 16-31);
    //     N = lane%16.
    const int f    = f0 + lo16;
    const float bias = bq[f];
#pragma unroll
    for (int v = 0; v < 8; ++v) {
        const int r = m0 + v + (hi << 3);
        out[(size_t)r * F_DIM + f] = acc[v] + bias;
    }
}

// ---------------------------------------------------------------------------
extern "C" void kernel_launch(void* const* d_in, const int* /*in_sizes*/, int /*n_in*/,
                              void* d_out, int /*out_size*/, void* /*d_ws*/,
                              size_t /*ws_size*/, hipStream_t stream) {
    const float* inputs     = (const float*)d_in[0];
    const int*   seq_len    = (const int*)  d_in[1];
    const int*   pack_idx   = (const int*)  d_in[2];
    const int*   masked_idx = (const int*)  d_in[3];
    const int*   neg_idx    = (const int*)  d_in[4];
    const float* mask_emb   = (const float*)d_in[5];
    const float* Wq         = (const float*)d_in[6];
    const float* bq         = (const float*)d_in[7];
    float* out = (float*)d_out;

    // 1) padded copy with zeroing (memory-bound bulk: ~268 MB streamed)
    k_copy_pad<<<N_B * C_DIM, 256, 0, stream>>>(inputs, seq_len, out);

    // 2) scatter the mask embeddings over the copied output (ordered by stream)
    k_scatter_mask<<<T_MASKED, 256, 0, stream>>>(pack_idx, masked_idx, mask_emb, out);

    // 3) gather-GEMM for unmasked (1024 rows) + negatives (10240 rows),
    //    written contiguously right after the big output.
    dim3 grid(M_ROWS / 16, F_DIM / 128);
    k_proj_gemm<<<grid, 256, 0, stream>>>(inputs, pack_idx, masked_idx, neg_idx,
                                          Wq, bq, out + OUT0_ELEMS);
}